// BezierRenderer_43224550868178
// MI455X (gfx1250) — compile-verified
//
#include <hip/hip_runtime.h>

typedef __attribute__((ext_vector_type(2))) float v2f;
typedef __attribute__((ext_vector_type(8))) float v8f;

#define GSZ 1024
#define NSTROKE 32
#define NSEG 16
#define SEG_STRIDE 10
#define META_STRIDE 12
#define SEG_FLOATS (NSTROKE * NSEG * SEG_STRIDE)   // 5120
#define META_OFF   SEG_FLOATS
#define META_FLOATS (NSTROKE * META_STRIDE)        // 384

// ---------------------------------------------------------------------------
// Kernel 1: one wave32. Bezier sampling as V_WMMA_F32_16X16X4_F32.
//   sample[row=(s,d)][p] = sum_k ctrl[row][k] * w_k(p), 64 rows -> 4 WMMAs.
// Then build per-segment and per-stroke render tables in workspace.
// ---------------------------------------------------------------------------
__global__ __launch_bounds__(32) void bezier_setup_kernel(
    const float* __restrict__ strokes,   // [32][2][4]
    const float* __restrict__ th,        // [32][1]
    const float* __restrict__ colors,    // [32][3]
    float* __restrict__ ws)
{
    __shared__ float pts[NSTROKE][17][2];   // polyline points * G

    const int  lane = threadIdx.x;
    const int  p    = lane & 15;
    const bool hi   = lane >= 16;

    // B fragment: Bezier basis, B[k][p], K=4, N=16.
    // VGPR0 = K0 (lanes 0-15) / K2 (lanes 16-31); VGPR1 = K1 / K3.
    const float tp  = (float)p;
    const float u   = 16.0f - tp;
    const float inv = 1.0f / 4096.0f;
    const float w0  = u * u * u * inv;
    const float w1  = 3.0f * u * u * tp * inv;
    const float w2  = 3.0f * u * tp * tp * inv;
    const float w3  = tp * tp * tp * inv;
    v2f b;
    b.x = hi ? w2 : w0;
    b.y = hi ? w3 : w1;

    for (int m = 0; m < 4; ++m) {
        // A fragment: row = (stroke,dim), control pts [p1 p2 p3 p4] along K.
        const int row = 16 * m + p;
        const int s   = row >> 1;
        const int d   = row & 1;
        const float* cs = strokes + s * 8;
        const float p0  = cs[d];
        const float de0 = cs[2 + d];
        const float p1v = cs[4 + d];
        const float de1 = cs[6 + d];
        const float c0 = p0;            // p1 = pts0
        const float c1 = p0 + de0;      // p2 = pts0 + deriv0
        const float c2 = p1v - de1;     // p3 = pts1 - deriv1
        const float c3 = p1v;           // p4 = pts1
        v2f a;
        a.x = hi ? c2 : c0;             // K0 / K2
        a.y = hi ? c3 : c1;             // K1 / K3

        v8f acc = {};
        acc = __builtin_amdgcn_wmma_f32_16x16x4_f32(
            /*neg_a=*/false, a, /*neg_b=*/false, b,
            /*c_mod=*/(short)0, acc, /*reuse_a=*/false, /*reuse_b=*/false);

        // Scatter D: VGPR i holds row (16m+i) lanes 0-15, row (16m+i+8) lanes 16-31.
        #pragma unroll
        for (int i = 0; i < 8; ++i) {
            const int rr = 16 * m + i + (hi ? 8 : 0);
            const int ss = rr >> 1;
            const int dd = rr & 1;
            pts[ss][p][dd] = acc[i] * (float)GSZ;
        }
        pts[s][16][d] = p1v * (float)GSZ;   // appended final point
    }
    __syncthreads();

    // Per-stroke metadata (one stroke per lane).
    const int s = lane;
    const float tval  = fmaxf(th[s] * 2.0f + 0.5f, 0.5f);
    const float r     = 2.0f * tval;          // falloff radius: darkness=0 beyond r
    const float inv2t = 1.0f / r;
    float xmin = 1e30f, xmax = -1e30f, ymin = 1e30f, ymax = -1e30f;
    #pragma unroll
    for (int k = 0; k < 17; ++k) {
        const float x = pts[s][k][0], y = pts[s][k][1];
        xmin = fminf(xmin, x); xmax = fmaxf(xmax, x);
        ymin = fminf(ymin, y); ymax = fmaxf(ymax, y);
    }
    float* M = ws + META_OFF + s * META_STRIDE;
    M[0] = inv2t;
    M[1] = r;
    M[2] = fminf(fmaxf(colors[s * 3 + 0], 0.0f), 1.0f);
    M[3] = fminf(fmaxf(colors[s * 3 + 1], 0.0f), 1.0f);
    M[4] = fminf(fmaxf(colors[s * 3 + 2], 0.0f), 1.0f);
    M[5] = xmin - r;  M[6] = xmax + r;
    M[7] = ymin - r;  M[8] = ymax + r;
    M[9] = 0.0f; M[10] = 0.0f; M[11] = 0.0f;

    // Per-segment table: {vx, vy, dx, dy, 1/(d2+eps), r-expanded bbox}.
    for (int k = 0; k < NSEG; ++k) {
        const float vx = pts[s][k][0],     vy = pts[s][k][1];
        const float wx = pts[s][k + 1][0], wy = pts[s][k + 1][1];
        const float dx = wx - vx, dy = wy - vy;
        const float invd2 = 1.0f / (dx * dx + dy * dy + 1e-5f);
        float* S = ws + (s * NSEG + k) * SEG_STRIDE;
        S[0] = vx; S[1] = vy; S[2] = dx; S[3] = dy; S[4] = invd2;
        S[5] = fminf(vx, wx) - r;  S[6] = fmaxf(vx, wx) + r;
        S[7] = fminf(vy, wy) - r;  S[8] = fmaxf(vy, wy) + r;
        S[9] = 0.0f;
    }
}

// ---------------------------------------------------------------------------
// Kernel 2: one 16x16 pixel tile per block. Segment table staged in LDS,
// uniform per-tile bbox culling (whole-wave skips), fused 3-channel max.
// ---------------------------------------------------------------------------
__global__ __launch_bounds__(256) void bezier_render_kernel(
    const float* __restrict__ ws, float* __restrict__ out)
{
    __shared__ float sseg[SEG_FLOATS];
    __shared__ float smeta[META_FLOATS];

    const int tid = threadIdx.y * 16 + threadIdx.x;
    for (int idx = tid; idx < SEG_FLOATS; idx += 256)  sseg[idx]  = ws[idx];
    for (int idx = tid; idx < META_FLOATS; idx += 256) smeta[idx] = ws[META_OFF + idx];
    __syncthreads();

    const int i = blockIdx.y * 16 + threadIdx.y;   // coords[...,0] = xg = i
    const int j = blockIdx.x * 16 + threadIdx.x;   // coords[...,1] = yg = j
    const float px = (float)i, py = (float)j;
    const float ti0 = (float)(blockIdx.y * 16), ti1 = ti0 + 15.0f;
    const float tj0 = (float)(blockIdx.x * 16), tj1 = tj0 + 15.0f;

    float r0 = 0.0f, r1 = 0.0f, r2 = 0.0f;
    for (int s = 0; s < NSTROKE; ++s) {
        const float* M = smeta + s * META_STRIDE;
        // Stroke-level cull (uniform across block -> cheap wave skip).
        if (M[5] > ti1 || M[6] < ti0 || M[7] > tj1 || M[8] < tj0) continue;
        const float inv2t = M[0];
        const float* SB = sseg + s * NSEG * SEG_STRIDE;
        float maxd = 0.0f;
        for (int k = 0; k < NSEG; ++k) {
            const float* S = SB + k * SEG_STRIDE;
            // Segment-level cull (also uniform).
            if (S[5] > ti1 || S[6] < ti0 || S[7] > tj1 || S[8] < tj0) continue;
            const float qx = px - S[0], qy = py - S[1];
            const float dx = S[2], dy = S[3];
            float tt = (qx * dx + qy * dy) * S[4];
            tt = fminf(fmaxf(tt, 0.0f), 1.0f);
            const float ex = qx - tt * dx;
            const float ey = qy - tt * dy;
            const float dist = sqrtf(ex * ex + ey * ey);
            maxd = fmaxf(maxd, 1.0f - dist * inv2t);   // (2t-dist)/(2t)
        }
        maxd = fminf(fmaxf(maxd, 0.0f), 1.0f);
        r0 = fmaxf(r0, maxd * M[2]);
        r1 = fmaxf(r1, maxd * M[3]);
        r2 = fmaxf(r2, maxd * M[4]);
    }

    const int base = i * GSZ + j;                  // out[c][i][j]
    out[base]               = r0;
    out[GSZ * GSZ + base]   = r1;
    out[2 * GSZ * GSZ + base] = r2;
}

// ---------------------------------------------------------------------------
extern "C" void kernel_launch(void* const* d_in, const int* in_sizes, int n_in,
                              void* d_out, int out_size, void* d_ws, size_t ws_size,
                              hipStream_t stream) {
    const float* strokes = (const float*)d_in[0];   // (32, 2, 4) f32
    const float* th      = (const float*)d_in[1];   // (32, 1)  f32
    const float* colors  = (const float*)d_in[2];   // (32, 3)  f32
    float* ws  = (float*)d_ws;                      // needs ~22 KB
    float* out = (float*)d_out;                     // (3, 1024, 1024) f32

    bezier_setup_kernel<<<1, 32, 0, stream>>>(strokes, th, colors, ws);
    bezier_render_kernel<<<dim3(GSZ / 16, GSZ / 16), dim3(16, 16), 0, stream>>>(ws, out);
}